// FilterNet_62053687493262
// MI455X (gfx1250) — compile-verified
//
#include <hip/hip_runtime.h>
#include <stdint.h>

// FilterNet on MI455X (gfx1250).
// Roofline: x = 2048*2048*16 fp32 = 256 MB must be read once -> ~11 us at
// 23.3 TB/s. All convs+adaptive-pools collapse algebraically to prefix-sum
// differences at 87 precomputed boundary positions per (batch, channel),
// so compute is ~1 add per loaded element: purely HBM-bound. WMMA cannot
// reduce the mandatory traffic; the CDNA5-idiomatic optimization is TDM
// (tensor_load_to_lds) bulk async staging of the strided stream into LDS,
// double-buffered and pipelined with s_wait_tensorcnt.

#define T_LEN   2048
#define F_CH    16
#define NSEG    64                         // time segments per batch row
#define SEGLEN  32                         // timesteps per segment
#define CHUNK_T 2                          // timesteps per TDM chunk
#define NCHUNK  (SEGLEN / CHUNK_T)         // 16 chunks
#define CHUNK_FLOATS (NSEG * CHUNK_T * F_CH) // 2048 floats = 8 KB per chunk
#define NU_MAX  512                        // >= unique boundary positions (<=504)

typedef unsigned int u32x4 __attribute__((ext_vector_type(4)));
typedef int          i32x4 __attribute__((ext_vector_type(4)));
typedef int          i32x8 __attribute__((ext_vector_type(8)));

#ifndef USE_TDM
#if defined(__has_builtin)
#if __has_builtin(__builtin_amdgcn_tensor_load_to_lds) && \
    __has_builtin(__builtin_amdgcn_s_wait_tensorcnt)
#define USE_TDM 1
#else
#define USE_TDM 0
#endif
#else
#define USE_TDM 0
#endif
#endif

struct FNTab {
  int nUniq;
  int segFirst[NSEG + 1];      // segFirst[w] = #unique positions <= 32*w
  short sIdx[6][12][6];        // rank of position s_{k,i}+j
  short eIdx[6][12][6];        // rank of position e_{k,i}+j
  short nA[6][12];             // stage-a bin widths
  unsigned short uniq[NU_MAX]; // sorted unique boundary positions
};

// ---------------------------------------------------------------------------
// Kernel 0: build boundary tables (tiny, deterministic, graph-capture safe).
// ---------------------------------------------------------------------------
__global__ void fn_build_tab(FNTab* __restrict__ tab) {
  __shared__ short rank[T_LEN + 1];
  if (threadIdx.x != 0) return;
  for (int p = 0; p <= T_LEN; ++p) rank[p] = 0;
  for (int k = 0; k < 6; ++k) {
    const int K = k + 1, L = T_LEN - k;          // conv-a output length 2049-K
    for (int i = 0; i < 12; ++i) {
      const int s = (i * L) / 12;
      const int e = ((i + 1) * L + 11) / 12;     // ceil
      tab->nA[k][i] = (short)(e - s);
      for (int j = 0; j < K; ++j) { rank[s + j] = 1; rank[e + j] = 1; }
    }
  }
  int n = 0;
  for (int p = 0; p <= T_LEN; ++p) {
    if (rank[p]) { tab->uniq[n] = (unsigned short)p; rank[p] = (short)n; ++n; }
    else rank[p] = -1;
  }
  tab->nUniq = n;
  for (int k = 0; k < 6; ++k) {
    const int K = k + 1, L = T_LEN - k;
    for (int i = 0; i < 12; ++i) {
      const int s = (i * L) / 12;
      const int e = ((i + 1) * L + 11) / 12;
      for (int j = 0; j < K; ++j) {
        tab->sIdx[k][i][j] = rank[s + j];
        tab->eIdx[k][i][j] = rank[e + j];
      }
    }
  }
  for (int w = 0; w <= NSEG; ++w) {
    int cnt = 0;
    for (int q = 0; q < n; ++q)
      if ((int)tab->uniq[q] <= SEGLEN * w) ++cnt;
    tab->segFirst[w] = cnt;
  }
}

#if USE_TDM
// Issue one TDM 2-D tile load: dim0 = CHUNK_T*F_CH fp32 contiguous (128 B),
// dim1 = NSEG lines with stride SEGLEN*F_CH elements (2 KB), packed to LDS.
// D# layout per CDNA5 ISA.
// This toolchain's builtin is the 6-arg form:
//   (u32x4 group0, i32x8 group1, i32x4 group2, i32x4 group3,
//    i32x8 group4/VADDR4 (unused -> zeros), i32 cpol)
__device__ __forceinline__ void tdm_issue_chunk(unsigned long long gaddr,
                                                unsigned int ldsaddr) {
  u32x4 g0;
  g0[0] = 1u;                                            // count=1, user mode
  g0[1] = ldsaddr;                                       // lds_addr (bytes)
  g0[2] = (unsigned int)(gaddr & 0xffffffffull);         // global_addr[31:0]
  g0[3] = (unsigned int)((gaddr >> 32) & 0x01ffffffull)  // global_addr[56:32]
        | (2u << 30);                                    // type=2 ("image")
  i32x8 g1;
  g1[0] = (int)(2u << 16);                               // data_size=2 (4 B)
  g1[1] = (int)((unsigned)(CHUNK_T * F_CH) << 16);       // tensor_dim0=32
  g1[2] = (int)((unsigned)NSEG << 16);                   // tensor_dim1=64
  g1[3] = (int)((unsigned)(CHUNK_T * F_CH) << 16);       // tile_dim0=32
  g1[4] = (int)(unsigned)NSEG;                           // tile_dim1=64, tile_dim2=0
  g1[5] = (int)(unsigned)(SEGLEN * F_CH);                // dim0_stride=512 elems
  g1[6] = 0;                                             // dim1_stride (unused, 2D)
  g1[7] = 0;
  i32x4 gz4 = {0, 0, 0, 0};                              // groups 2/3: 2-D tensor
  i32x8 gz8 = {0, 0, 0, 0, 0, 0, 0, 0};                  // VADDR4: unused
  __builtin_amdgcn_tensor_load_to_lds(g0, g1, gz4, gz4, gz8, 0);
}
#endif

// ---------------------------------------------------------------------------
// Kernel 1: one 256-thread block per batch row. Single streaming pass over x.
// Thread (c4 = tid&3, w = tid>>2) owns channels [4c4,4c4+3] of segment w.
// ---------------------------------------------------------------------------
__global__ __launch_bounds__(256) void fn_main(
    const float* __restrict__ x,
    const float* __restrict__ wa0, const float* __restrict__ wa1,
    const float* __restrict__ wa2, const float* __restrict__ wa3,
    const float* __restrict__ wa4, const float* __restrict__ wa5,
    const float* __restrict__ wb0, const float* __restrict__ wb1,
    const float* __restrict__ wb2, const float* __restrict__ wb3,
    const float* __restrict__ wb4, const float* __restrict__ wb5,
    const float* __restrict__ wfc,
    float* __restrict__ out,
    const FNTab* __restrict__ tab)
{
  __shared__ float P[NU_MAX][F_CH];        // prefix sums at boundary positions
#if USE_TDM
  __shared__ float buf[2][CHUNK_FLOATS];   // TDM double buffer (2 x 8 KB)
#endif
  __shared__ float segtot[NSEG][F_CH];     // per-segment channel totals
  __shared__ unsigned short uniqL[NU_MAX]; // sorted unique positions
  __shared__ float hL[112];                // 108 features (k*18 + r layout)

  const int tid = threadIdx.x;
  const int b   = blockIdx.x;
  const int nU  = tab->nUniq;
  for (int q = tid; q < nU; q += 256) uniqL[q] = tab->uniq[q];

  const int c4    = tid & 3;
  const int w     = tid >> 2;
  const int evBeg = tab->segFirst[w];
  const int evEnd = tab->segFirst[w + 1];
  if (tid < F_CH && tab->segFirst[0] > 0) P[0][tid] = 0.0f; // prefix at pos 0
#if USE_TDM
  const unsigned long long gband =
      (unsigned long long)(uintptr_t)(x + (size_t)b * (T_LEN * F_CH));
  // Low 32 bits of a flat shared pointer are the LDS byte address
  // (LDS aperture lives entirely in addr[63:32] per ISA aperture check).
  const unsigned int lds0 = (unsigned int)(uintptr_t)&buf[0][0];
  const unsigned int lds1 = (unsigned int)(uintptr_t)&buf[1][0];
  if (tid < 32) {                          // wave 0 drives the TDM pipeline
    tdm_issue_chunk(gband + 0ull * (CHUNK_T * F_CH * 4), lds0);
    tdm_issue_chunk(gband + 1ull * (CHUNK_T * F_CH * 4), lds1);
  }
#endif
  __syncthreads();

  // ---- Phase 1: stream the row, running sums in registers ------------------
  float rx = 0.f, ry = 0.f, rz = 0.f, rw_ = 0.f;
  int ev = evBeg;
  int nextPos = (ev < evEnd) ? (int)uniqL[ev] : 0x7fffffff;
  const int tbase = w * SEGLEN;

#if USE_TDM
  for (int cc = 0; cc < NCHUNK; ++cc) {
    if (tid < 32) {
      if (cc < NCHUNK - 1) __builtin_amdgcn_s_wait_tensorcnt(1); // chunk cc done
      else                 __builtin_amdgcn_s_wait_tensorcnt(0);
    }
    __syncthreads();                       // TDM data visible to all waves
    const float* src = &buf[cc & 1][w * (CHUNK_T * F_CH) + c4 * 4];
#pragma unroll
    for (int tl = 0; tl < CHUNK_T; ++tl) {
      const float4 v = *(const float4*)(src + tl * F_CH);
      rx += v.x; ry += v.y; rz += v.z; rw_ += v.w;
      if (tbase + cc * CHUNK_T + tl + 1 == nextPos) {
        float* row = &P[ev][c4 * 4];
        row[0] = rx; row[1] = ry; row[2] = rz; row[3] = rw_;
        ++ev;
        nextPos = (ev < evEnd) ? (int)uniqL[ev] : 0x7fffffff;
      }
    }
    __syncthreads();                       // buffer fully consumed
    if (tid < 32 && cc + 2 < NCHUNK)
      tdm_issue_chunk(gband + (unsigned long long)(cc + 2) * (CHUNK_T * F_CH * 4),
                      (cc & 1) ? lds1 : lds0);
  }
#else
  const float4* gp =
      (const float4*)(x + (size_t)b * (T_LEN * F_CH) + (size_t)w * SEGLEN * F_CH) + c4;
  for (int u = 0; u < SEGLEN; ++u) {
    const float4 v = gp[u * 4];
    __builtin_prefetch((const char*)(gp + u * 4) + 2048, 0, 0); // global_prefetch_b8
    rx += v.x; ry += v.y; rz += v.z; rw_ += v.w;
    if (tbase + u + 1 == nextPos) {
      float* row = &P[ev][c4 * 4];
      row[0] = rx; row[1] = ry; row[2] = rz; row[3] = rw_;
      ++ev;
      nextPos = (ev < evEnd) ? (int)uniqL[ev] : 0x7fffffff;
    }
  }
#endif
  { float* row = &segtot[w][c4 * 4]; row[0] = rx; row[1] = ry; row[2] = rz; row[3] = rw_; }
  __syncthreads();

  // ---- Phase 2: segment-exclusive scan, promote partials to true prefixes --
  float px = 0.f, py = 0.f, pz = 0.f, pw = 0.f;
  for (int w2 = 0; w2 < w; ++w2) {
    const float* row = &segtot[w2][c4 * 4];
    px += row[0]; py += row[1]; pz += row[2]; pw += row[3];
  }
  for (int q = evBeg; q < evEnd; ++q) {
    float* row = &P[q][c4 * 4];
    row[0] += px; row[1] += py; row[2] += pz; row[3] += pw;
  }
  __syncthreads();

  // ---- Phase 3: 72 pooled stage-a features = wa . (prefix differences) -----
  if (tid < 72) {
    const float* waArr[6] = {wa0, wa1, wa2, wa3, wa4, wa5};
    const int k = tid / 12, i = tid % 12, K = k + 1;
    const float* wa = waArr[k];
    float acc = 0.0f;
    for (int j = 0; j < K; ++j) {
      const int si = tab->sIdx[k][i][j];
      const int ei = tab->eIdx[k][i][j];
      const float* Ps = P[si];
      const float* Pe = P[ei];
      for (int c = 0; c < F_CH; ++c)
        acc += wa[c * K + j] * (Pe[c] - Ps[c]);
    }
    const float v = acc / (float)tab->nA[k][i];
    hL[k * 18 + i] = (v > 0.0f) ? v : 0.1f * v;     // leaky relu 0.1
  }
  __syncthreads();

  // ---- Phase 4: stage-b dilated conv (dil=2) + pool->6 + leaky -------------
  if (tid < 6) {
    const float* wbArr[6] = {wb0, wb1, wb2, wb3, wb4, wb5};
    const int k = tid, K = k + 1, lb = 12 - 2 * k;
    const float* wb = wbArr[k];
    float fb[12];
    for (int l = 0; l < lb; ++l) {
      float s = 0.0f;
      for (int j = 0; j < K; ++j) s += hL[k * 18 + l + 2 * j] * wb[j];
      fb[l] = s;
    }
    for (int i = 0; i < 6; ++i) {
      const int s = (i * lb) / 6;
      const int e = ((i + 1) * lb + 5) / 6;
      float a = 0.0f;
      for (int t = s; t < e; ++t) a += fb[t];
      a /= (float)(e - s);
      hL[k * 18 + 12 + i] = (a > 0.0f) ? a : 0.1f * a;
    }
  }
  __syncthreads();

  // ---- Phase 5: final 108-dot ---------------------------------------------
  if (tid == 0) {
    float acc = 0.0f;
    for (int q = 0; q < 108; ++q) acc += hL[q] * wfc[q];
    out[b] = acc;
  }
}

// ---------------------------------------------------------------------------
extern "C" void kernel_launch(void* const* d_in, const int* in_sizes, int n_in,
                              void* d_out, int out_size, void* d_ws, size_t ws_size,
                              hipStream_t stream) {
  (void)in_sizes; (void)n_in; (void)ws_size;
  // setup_inputs() dict order: x, wa1, wb1, wa2, wb2, ..., wa6, wb6, wfc
  const float* x = (const float*)d_in[0];
  const float* wa[6];
  const float* wb[6];
  for (int k = 0; k < 6; ++k) {
    wa[k] = (const float*)d_in[1 + 2 * k];
    wb[k] = (const float*)d_in[2 + 2 * k];
  }
  const float* wfc = (const float*)d_in[13];
  float* out = (float*)d_out;
  FNTab* tab = (FNTab*)d_ws;

  fn_build_tab<<<1, 1, 0, stream>>>(tab);
  fn_main<<<out_size, 256, 0, stream>>>(
      x, wa[0], wa[1], wa[2], wa[3], wa[4], wa[5],
      wb[0], wb[1], wb[2], wb[3], wb[4], wb[5],
      wfc, out, tab);
}